// GraphConvResBlock_69217692942601
// MI455X (gfx1250) — compile-verified
//
#include <hip/hip_runtime.h>
#include <math.h>

// ---------------------------------------------------------------------------
// GraphConv residual block for MI455X (gfx1250), fp32 WMMA (16x16x4) path.
// Memory-bound (~93MB HBM after L2 reuse, ~8 GFLOP) -> keep fp32 accuracy,
// run all matmuls on the matrix pipe via V_WMMA_F32_16X16X4_F32.
// Each wave computes a 32x16 output supertile (2 independent accumulators)
// so B-fragments are amortized and the WMMA chains pipeline without the
// per-instruction D->C dependency. Next-chunk rows are prefetched with
// global_prefetch_b8.
// ---------------------------------------------------------------------------

typedef __attribute__((ext_vector_type(2))) float v2f;
typedef __attribute__((ext_vector_type(8))) float v8f;

#define NGRAPH 64
#define LPADMAX 512
#define CIN 128
#define BN_EPS 1e-5f

static __device__ __forceinline__ v8f wmma_f32(v2f a, v2f b, v8f c) {
  return __builtin_amdgcn_wmma_f32_16x16x4_f32(false, a, false, b,
                                               (short)0, c, false, false);
}

// ---------------- tiny setup kernels ----------------------------------------

__global__ void prefix_off_kernel(const int* __restrict__ ld, int* __restrict__ off) {
  if (threadIdx.x == 0 && blockIdx.x == 0) {
    int s = 0;
    for (int i = 0; i < NGRAPH; ++i) { off[i] = s; s += ld[i]; }
    off[NGRAPH] = s;
  }
}

// Zero-pad ragged x [M,128] into Xpad [B,512,128]; build node -> padded-row map.
__global__ __launch_bounds__(256) void scatter_pad_kernel(
    const float* __restrict__ x, const int* __restrict__ ld,
    const int* __restrict__ off, float* __restrict__ Xp,
    int* __restrict__ nodeRow) {
  int b = blockIdx.x;
  int l = ld[b];
  int o = off[b];
  const int elems = LPADMAX * CIN;
  float* dst = Xp + (size_t)b * elems;
  for (int i = threadIdx.x; i < elems; i += blockDim.x) {
    int pos = i >> 7;               // /CIN
    int c = i & (CIN - 1);
    dst[i] = (pos < l) ? x[(size_t)(o + pos) * CIN + c] : 0.0f;
  }
  for (int pos = threadIdx.x; pos < l; pos += blockDim.x)
    nodeRow[o + pos] = b * LPADMAX + pos;
}

// ---------------- aggregate: Hp[b] = am[b] @ Xp[b] ---------------------------
// One wave -> 32x16 C-supertile (2 acc chains); 8 waves/block -> all 8 column
// tiles of N=128. K clipped to ceil16(l) (padded rows of Xp are zero -> exact).

__global__ __launch_bounds__(256) void aggregate_wmma_kernel(
    const float* __restrict__ am, const float* __restrict__ Xp,
    float* __restrict__ Hp, const int* __restrict__ ld) {
  int rt = blockIdx.x;                 // 32-row supertile (0..15)
  int b  = blockIdx.y;                 // graph
  int l = ld[b];
  int lpad = (l + 15) & ~15;
  int row0 = rt * 32;
  if (row0 >= lpad) return;            // block-uniform: EXEC stays full

  int tid  = threadIdx.x;
  int lane = tid & 31;
  int wv   = tid >> 5;                 // 0..7 -> column tile
  int n0   = wv * 16;
  int mn   = lane & 15;                // m for A-frag, n for B/C-frag
  int kh   = lane >> 4;                // k-half (0 or 1)

  const float* A0 = am + (size_t)b * LPADMAX * LPADMAX + (size_t)(row0 + mn) * LPADMAX;
  const float* A1 = A0 + (size_t)16 * LPADMAX;
  const float* Bm = Xp + (size_t)b * LPADMAX * CIN + n0 + mn;
  float* C = Hp + (size_t)b * LPADMAX * CIN;

  v8f acc0 = {}, acc1 = {};
  for (int kc = 0; kc < lpad; kc += 16) {
    if (kc + 16 < lpad) {              // prefetch next 64B line per tile row
      __builtin_prefetch(A0 + kc + 16, 0, 3);
      __builtin_prefetch(A1 + kc + 16, 0, 3);
      __builtin_prefetch(Bm + (size_t)(kc + 16) * CIN, 0, 3);
    }
#pragma unroll
    for (int s = 0; s < 4; ++s) {
      int kk = kc + s * 4 + kh * 2;
      v2f bb;                          // B 4x16 frag: B[kk][n], B[kk+1][n]
      bb.x = Bm[(size_t)kk * CIN];
      bb.y = Bm[(size_t)(kk + 1) * CIN];
      v2f a0 = *(const v2f*)(A0 + kk); // A 16x4 frags (kk even -> 8B aligned)
      v2f a1 = *(const v2f*)(A1 + kk);
      acc0 = wmma_f32(a0, bb, acc0);   // independent chains -> pipelined
      acc1 = wmma_f32(a1, bb, acc1);
    }
  }
#pragma unroll
  for (int r = 0; r < 8; ++r) {
    C[(size_t)(row0 + kh * 8 + r) * CIN + n0 + mn]      = acc0[r];
    C[(size_t)(row0 + 16 + kh * 8 + r) * CIN + n0 + mn] = acc1[r];
  }
}

// ---------------- fused linear (x @ W^T + bias) + BN (+activation) -----------
// MODE 0: padded rows, ReLU, zero rows with pos >= l (aux = ld), COUT=128
// MODE 1: ragged rows, no activation (residual branch),           COUT=256
// MODE 2: A-rows gathered via aux=nodeRow, += res, leaky-ReLU,    COUT=256
// Each wave computes a 32x16 supertile; W-fragment shared by both acc chains.

template <int MODE, int COUT>
__global__ __launch_bounds__(256) void linbn_wmma_kernel(
    const float* __restrict__ Xin, const float* __restrict__ W,
    const float* __restrict__ bias, const float* __restrict__ gamma,
    const float* __restrict__ beta, const float* __restrict__ mean,
    const float* __restrict__ var, const float* __restrict__ res,
    const int* __restrict__ aux, float* __restrict__ Y) {
  int rt   = blockIdx.x;
  int tid  = threadIdx.x;
  int lane = tid & 31;
  int wv   = tid >> 5;
  int ct   = blockIdx.y * 8 + wv;
  int row0 = rt * 32;
  int n0   = ct * 16;
  int mn   = lane & 15;
  int kh   = lane >> 4;

  size_t asrc0, asrc1;
  if (MODE == 2) {                     // gather padded rows
    asrc0 = (size_t)aux[row0 + mn] * CIN;
    asrc1 = (size_t)aux[row0 + 16 + mn] * CIN;
  } else {
    asrc0 = (size_t)(row0 + mn) * CIN;
    asrc1 = (size_t)(row0 + 16 + mn) * CIN;
  }

  const float* Wr = W + (size_t)(n0 + mn) * CIN;        // B'[k][n] = W[n][k]

  v8f acc0 = {}, acc1 = {};
#pragma unroll
  for (int t = 0; t < (CIN >> 2); ++t) {
    int kk = t * 4 + kh * 2;
    v2f bb = *(const v2f*)(Wr + kk);
    v2f a0 = *(const v2f*)(Xin + asrc0 + kk);
    v2f a1 = *(const v2f*)(Xin + asrc1 + kk);
    acc0 = wmma_f32(a0, bb, acc0);
    acc1 = wmma_f32(a1, bb, acc1);
  }

  // Fused BN epilogue: ((v + bias) - mean) * gamma/sqrt(var+eps) + beta
  int n = n0 + mn;
  float s  = gamma[n] / sqrtf(var[n] + BN_EPS);
  float sh = beta[n] - mean[n] * s;
  float bi = bias[n];
  int l = 0;
  if (MODE == 0) l = aux[row0 >> 9];   // 32-row tile never straddles a graph

#pragma unroll
  for (int half = 0; half < 2; ++half) {
    v8f acc = half ? acc1 : acc0;
#pragma unroll
    for (int r = 0; r < 8; ++r) {
      int row = row0 + half * 16 + kh * 8 + r;
      float v = (acc[r] + bi) * s + sh;
      if (MODE == 0) {
        v = v > 0.0f ? v : 0.0f;                       // ReLU
        if ((row & (LPADMAX - 1)) >= l) v = 0.0f;      // zero padding rows
      } else if (MODE == 2) {
        v += res[(size_t)row * COUT + n];              // residual add
        v = v > 0.0f ? v : 0.01f * v;                  // leaky ReLU
      }
      Y[(size_t)row * COUT + n] = v;
    }
  }
}

// ---------------- launch -----------------------------------------------------

extern "C" void kernel_launch(void* const* d_in, const int* in_sizes, int n_in,
                              void* d_out, int out_size, void* d_ws, size_t ws_size,
                              hipStream_t stream) {
  const float* x    = (const float*)d_in[0];
  const int*   ld   = (const int*)d_in[1];
  const float* am   = (const float*)d_in[2];
  const float* Wl   = (const float*)d_in[3];
  const float* bl   = (const float*)d_in[4];
  const float* gl   = (const float*)d_in[5];
  const float* betal= (const float*)d_in[6];
  const float* ml   = (const float*)d_in[7];
  const float* vl   = (const float*)d_in[8];
  const float* W1   = (const float*)d_in[9];
  const float* b1   = (const float*)d_in[10];
  const float* g1   = (const float*)d_in[11];
  const float* beta1= (const float*)d_in[12];
  const float* m1   = (const float*)d_in[13];
  const float* v1   = (const float*)d_in[14];
  const float* W2   = (const float*)d_in[15];
  const float* b2   = (const float*)d_in[16];
  const float* g2   = (const float*)d_in[17];
  const float* beta2= (const float*)d_in[18];
  const float* m2   = (const float*)d_in[19];
  const float* v2   = (const float*)d_in[20];
  float* out = (float*)d_out;

  const int M = in_sizes[0] / CIN;                       // 16384
  const size_t PADF = (size_t)NGRAPH * LPADMAX * CIN;    // 4M floats = 16 MiB

  char* ws = (char*)d_ws;
  int*   off     = (int*)ws;                             // 65 ints
  int*   nodeRow = (int*)(ws + 1024);                    // M ints
  float* Xpad    = (float*)(ws + (1 << 17));             // 16 MiB (reused as Gpad)
  float* H1      = Xpad + PADF;                          // 16 MiB (reused as Res)
  float* H2      = H1 + PADF;                            // 16 MiB

  // 1) per-graph offsets
  hipLaunchKernelGGL(prefix_off_kernel, dim3(1), dim3(32), 0, stream, ld, off);
  // 2) zero-pad x, build gather map
  hipLaunchKernelGGL(scatter_pad_kernel, dim3(NGRAPH), dim3(256), 0, stream,
                     x, ld, off, Xpad, nodeRow);
  // 3) aggregate1: H1 = am @ Xpad
  hipLaunchKernelGGL(aggregate_wmma_kernel, dim3(LPADMAX / 32, NGRAPH), dim3(256), 0, stream,
                     am, Xpad, H1, ld);
  // 4) conv1 + BN + ReLU (+ zero pad rows): H2 = act(H1 @ W1^T)
  hipLaunchKernelGGL((linbn_wmma_kernel<0, 128>),
                     dim3(NGRAPH * LPADMAX / 32, 1), dim3(256), 0, stream,
                     H1, W1, b1, g1, beta1, m1, v1, nullptr, ld, H2);
  // 5) aggregate2: Gpad(=Xpad) = am @ H2   (am now L2-resident)
  hipLaunchKernelGGL(aggregate_wmma_kernel, dim3(LPADMAX / 32, NGRAPH), dim3(256), 0, stream,
                     am, H2, Xpad, ld);
  // 6) residual branch: Res(=H1) = BN(x @ Wl^T)
  hipLaunchKernelGGL((linbn_wmma_kernel<1, 256>),
                     dim3(M / 32, 2), dim3(256), 0, stream,
                     x, Wl, bl, gl, betal, ml, vl, nullptr, nullptr, H1);
  // 7) conv2 + BN + residual + leaky-ReLU, gathered -> d_out
  hipLaunchKernelGGL((linbn_wmma_kernel<2, 256>),
                     dim3(M / 32, 2), dim3(256), 0, stream,
                     Xpad, W2, b2, g2, beta2, m2, v2, H1, nodeRow, out);

  (void)n_in; (void)out_size; (void)ws_size;
}